// STBlock_10471130267991
// MI455X (gfx1250) — compile-verified
//
#include <hip/hip_runtime.h>
#include <hip/hip_bf16.h>

// ---------------------------------------------------------------------------
// Spatio-temporal graph block for MI455X (gfx1250, wave32, WMMA + async-LDS).
//   B=1, T=32, H=128, heads=4, D=32, ED=16, conv K=3. N and E derived at launch.
// ---------------------------------------------------------------------------

typedef __attribute__((ext_vector_type(2))) float v2f;
typedef __attribute__((ext_vector_type(8))) float v8f;

#define TT 32
#define HH 128
#define LP 132            // padded LDS row stride (floats): 528B = 16B-aligned
#define NEG_SLOPE 0.2f

// ordered-uint encoding so unsigned atomicMax == float max
__device__ __forceinline__ unsigned f2ord(float f) {
    unsigned u = __float_as_uint(f);
    return (u & 0x80000000u) ? ~u : (u | 0x80000000u);
}
__device__ __forceinline__ float ord2f(unsigned u) {
    return (u & 0x80000000u) ? __uint_as_float(u & 0x7fffffffu) : __uint_as_float(~u);
}

// async DMA: 16 bytes global -> LDS (ASYNCcnt-tracked, no VGPR data path)
__device__ __forceinline__ void async_copy16(unsigned lds_addr, const float* gsrc) {
    asm volatile("global_load_async_to_lds_b128 %0, %1, off"
                 :: "v"(lds_addr), "v"(gsrc) : "memory");
}
__device__ __forceinline__ void async_wait0() {
    asm volatile("s_wait_asynccnt 0x0" ::: "memory");
}

// ---------------------------------------------------------------------------
// Kernel 1: per-node temporal Conv1d (K=3, SAME) via V_WMMA_F32_16X16X4_F32,
//           + bias + residual + LayerNorm1.  One block per node, 8 waves.
// GEMM view: Y[t,ho] = sum_k sum_hi Xpad[t+k-1,hi] * W[ho,hi,k]
// sX rows are time-shifted: sX[0] and sX[33] are zero pad rows (t = -1, 32).
// ---------------------------------------------------------------------------
__global__ __launch_bounds__(256) void k_conv_ln1(
    const float* __restrict__ x, const float* __restrict__ conv_w,
    const float* __restrict__ conv_b, const float* __restrict__ ln1_g,
    const float* __restrict__ ln1_b, float* __restrict__ x1)
{
    __shared__ float sX[TT + 2][LP];
    __shared__ float sY[TT][LP];

    const int n   = blockIdx.x;
    const int tid = threadIdx.x;
    const float* xn = x + (size_t)n * TT * HH;

    // zero-pad rows (only cols 0..127 are ever read)
    for (int i = tid; i < 2 * HH; i += 256)
        sX[(i < HH) ? 0 : (TT + 1)][i & 127] = 0.0f;

    // async DMA the 32x128 tile into sX rows 1..32 (16 floats / thread, 4x b128)
    {
        const int idx0 = tid * 16;               // 4096 floats total
        const int row  = idx0 >> 7;
        const int col  = idx0 & 127;
        const unsigned lds = (unsigned)(uintptr_t)&sX[row + 1][col];
        const float* src   = xn + idx0;
        async_copy16(lds,      src);
        async_copy16(lds + 16, src + 4);
        async_copy16(lds + 32, src + 8);
        async_copy16(lds + 48, src + 12);
    }
    async_wait0();
    __syncthreads();

    const int wave    = tid >> 5;
    const int lane    = tid & 31;
    const int halfsel = lane >> 4;   // 0: K0/K1 half, 1: K2/K3 half
    const int l15     = lane & 15;

    // 16 tiles of 16x16 (t0 in {0,16} x h0 in {0..112}); each wave does 2
    for (int tile = wave; tile < 16; tile += 8) {
        const int t0 = (tile >> 3) << 4;
        const int h0 = (tile & 7) << 4;
        v8f c = {};
        for (int k = 0; k < 3; ++k) {
            const int tt = t0 + l15 + k;         // padded row (time t0+l15+k-1)
            const float* wcol = conv_w + (size_t)(h0 + l15) * (HH * 3) + k;
            for (int hi0 = 0; hi0 < HH; hi0 += 4) {
                const int ra = hi0 + halfsel * 2;        // this half's K pair
                const v2f a = *(const v2f*)&sX[tt][ra];  // ds_load_b64
                v2f b;
                b.x = wcol[(size_t)ra * 3];              // B[r][n] = W[n, r, k]
                b.y = wcol[(size_t)(ra + 1) * 3];
                c = __builtin_amdgcn_wmma_f32_16x16x4_f32(
                        false, a, false, b, (short)0, c, false, false);
            }
        }
        const float bias = conv_b[h0 + l15];
        #pragma unroll
        for (int v = 0; v < 8; ++v)                   // C: VGPR v -> M=v / v+8
            sY[t0 + v + halfsel * 8][h0 + l15] = c[v] + bias;
    }
    __syncthreads();

    // LayerNorm over H per time row: 8 lanes per row, 16 elems per lane
    const int row = tid >> 3;
    const int sub = tid & 7;
    float r[16], s = 0.f, s2 = 0.f;
    #pragma unroll
    for (int i = 0; i < 16; ++i) {
        const int h = sub * 16 + i;
        const float val = sX[row + 1][h] + sY[row][h];
        r[i] = val; s += val; s2 += val * val;
    }
    #pragma unroll
    for (int off = 4; off >= 1; off >>= 1) {
        s  += __shfl_xor(s,  off, 32);
        s2 += __shfl_xor(s2, off, 32);
    }
    const float mean = s * (1.0f / HH);
    const float inv  = rsqrtf(s2 * (1.0f / HH) - mean * mean + 1e-5f);
    float* op = x1 + ((size_t)n * TT + row) * HH;
    #pragma unroll
    for (int i = 0; i < 16; ++i) {
        const int h = sub * 16 + i;
        op[h] = (r[i] - mean) * inv * ln1_g[h] + ln1_b[h];
    }
}

// ---------------------------------------------------------------------------
// Kernel 2: gl = xs@Wl, gr = xs@Wr via WMMA f32. Rows are (node*T + t) from x1;
// outputs stored as [g=t][node][H]. One block = 16 rows, 8 waves = 8 col tiles.
// ---------------------------------------------------------------------------
__global__ __launch_bounds__(256) void k_gat_transform(
    const float* __restrict__ x1, const float* __restrict__ Wl,
    const float* __restrict__ Wr, float* __restrict__ gl,
    float* __restrict__ gr, int Nn)
{
    __shared__ float sA[16][LP];
    const int r0  = blockIdx.x * 16;
    const int tid = threadIdx.x;

    // async DMA 16x128 tile (8 floats / thread, 2x b128)
    {
        const int idx0 = tid * 8;                // 2048 floats total
        const int row  = idx0 >> 7;
        const int col  = idx0 & 127;
        const unsigned lds = (unsigned)(uintptr_t)&sA[row][col];
        const float* src   = x1 + (size_t)r0 * HH + idx0;
        async_copy16(lds,      src);
        async_copy16(lds + 16, src + 4);
    }
    async_wait0();
    __syncthreads();

    const int lane    = tid & 31;
    const int h0      = (tid >> 5) * 16;
    const int halfsel = lane >> 4;
    const int l15     = lane & 15;
    const int nn      = h0 + l15;

    v8f cl = {}, cr = {};
    for (int k0 = 0; k0 < HH; k0 += 4) {
        const int ra = k0 + halfsel * 2;
        const v2f a = *(const v2f*)&sA[l15][ra];         // ds_load_b64
        v2f bl, br;
        bl.x = Wl[(size_t)ra * HH + nn];       bl.y = Wl[(size_t)(ra + 1) * HH + nn];
        br.x = Wr[(size_t)ra * HH + nn];       br.y = Wr[(size_t)(ra + 1) * HH + nn];
        cl = __builtin_amdgcn_wmma_f32_16x16x4_f32(false, a, false, bl, (short)0, cl, false, false);
        cr = __builtin_amdgcn_wmma_f32_16x16x4_f32(false, a, false, br, (short)0, cr, false, false);
    }
    #pragma unroll
    for (int v = 0; v < 8; ++v) {
        const int row  = r0 + v + halfsel * 8;   // row = node*32 + t
        const int node = row >> 5;
        const int t    = row & 31;
        const size_t o = ((size_t)t * Nn + node) * HH + nn;
        gl[o] = cl[v];
        gr[o] = cr[v];
    }
}

// ---------------------------------------------------------------------------
// Kernel 3: per-dst mean of incoming edge attrs (counts + sums via atomics)
// ---------------------------------------------------------------------------
__global__ void k_loop_sum(const int* __restrict__ ei,
                           const float* __restrict__ edge_attr,
                           float* __restrict__ esum, unsigned* __restrict__ cnt,
                           int E)
{
    const int i = blockIdx.x * blockDim.x + threadIdx.x;
    if (i >= E * 16) return;
    const int e = i >> 4, j = i & 15;
    const int d = ei[E + e];                 // edge_index[1][e]
    atomicAdd(&esum[d * 16 + j], edge_attr[e * 16 + j]);
    if (j == 0) atomicAdd(&cnt[d], 1u);
}

// ---------------------------------------------------------------------------
// Kernel 4: ee = concat(edge_attr, loop_attr) @ We    [(E+N) x 128]
// ---------------------------------------------------------------------------
__global__ void k_edge_transform(const float* __restrict__ edge_attr,
                                 const float* __restrict__ esum,
                                 const unsigned* __restrict__ cnt,
                                 const float* __restrict__ We,
                                 float* __restrict__ ee, int E, int Nn)
{
    const int i = blockIdx.x * blockDim.x + threadIdx.x;
    if (i >= (E + Nn) * HH) return;
    const int r = i >> 7, h = i & 127;
    float acc = 0.f;
    if (r < E) {
        const float* a = edge_attr + (size_t)r * 16;
        #pragma unroll
        for (int f = 0; f < 16; ++f) acc += a[f] * We[f * HH + h];
    } else {
        const int nd = r - E;
        const float ic = 1.0f / fmaxf((float)cnt[nd], 1.0f);
        const float* a = esum + (size_t)nd * 16;
        #pragma unroll
        for (int f = 0; f < 16; ++f) acc += (a[f] * ic) * We[f * HH + h];
    }
    ee[i] = acc;
}

// ---------------------------------------------------------------------------
// Kernel 5: GATv2 logits + scatter-max  (one thread per (edge, g, k))
// ---------------------------------------------------------------------------
__global__ void k_logits(const int* __restrict__ ei, const float* __restrict__ gl,
                         const float* __restrict__ gr, const float* __restrict__ ee,
                         const float* __restrict__ att, float* __restrict__ logits,
                         unsigned* __restrict__ mx, int E, int Nn)
{
    const int i = blockIdx.x * blockDim.x + threadIdx.x;
    if (i >= (E + Nn) * 128) return;
    const int k = i & 3, g = (i >> 2) & 31, e = i >> 7;
    const int src = (e < E) ? ei[e]     : (e - E);
    const int dst = (e < E) ? ei[E + e] : (e - E);
    const float* pl = gl + ((size_t)g * Nn + src) * HH + k * 32;
    const float* pr = gr + ((size_t)g * Nn + dst) * HH + k * 32;
    const float* pe = ee + (size_t)e * HH + k * 32;
    const float* pa = att + k * 32;
    float acc = 0.f;
    #pragma unroll 8
    for (int d = 0; d < 32; ++d) {
        float s = pl[d] + pr[d] + pe[d];
        s = (s > 0.f) ? s : NEG_SLOPE * s;
        acc += s * pa[d];
    }
    logits[i] = acc;
    atomicMax(&mx[((size_t)dst * 32 + g) * 4 + k], f2ord(acc));
}

// ---------------------------------------------------------------------------
// Kernel 6: ex = exp(logit - max); scatter-add denominator and e_src*ex
// ---------------------------------------------------------------------------
__global__ void k_aggregate(const int* __restrict__ ei, const float* __restrict__ gl,
                            const float* __restrict__ logits,
                            const unsigned* __restrict__ mx,
                            float* __restrict__ den, float* __restrict__ aggr,
                            int E, int Nn)
{
    const int i = blockIdx.x * blockDim.x + threadIdx.x;
    if (i >= (E + Nn) * 128) return;
    const int k = i & 3, g = (i >> 2) & 31, e = i >> 7;
    const int src = (e < E) ? ei[e]     : (e - E);
    const int dst = (e < E) ? ei[E + e] : (e - E);
    const float m  = ord2f(mx[((size_t)dst * 32 + g) * 4 + k]);
    const float ex = __expf(logits[i] - m);
    atomicAdd(&den[((size_t)dst * 32 + g) * 4 + k], ex);
    const float* pl = gl + ((size_t)g * Nn + src) * HH + k * 32;
    float* pa = aggr + ((size_t)dst * 32 + g) * HH + k * 32;
    #pragma unroll 8
    for (int d = 0; d < 32; ++d) atomicAdd(&pa[d], pl[d] * ex);
}

// ---------------------------------------------------------------------------
// Kernel 7: out = LN2(x1 + aggr/den + gat_b); 8 lanes per (n,t) row
// ---------------------------------------------------------------------------
__global__ __launch_bounds__(256) void k_final_ln2(
    const float* __restrict__ x1, const float* __restrict__ aggr,
    const float* __restrict__ den, const float* __restrict__ gat_b,
    const float* __restrict__ ln2_g, const float* __restrict__ ln2_b,
    float* __restrict__ out, int rows)
{
    const int gid = blockIdx.x * 256 + threadIdx.x;
    const int row = gid >> 3;
    const int sub = gid & 7;
    if (row >= rows) return;
    const float* px = x1   + (size_t)row * HH;
    const float* pa = aggr + (size_t)row * HH;   // row = n*32 + t == (dst*32 + g)
    const float* pd = den  + (size_t)row * 4;
    float r[16], s = 0.f, s2 = 0.f;
    #pragma unroll
    for (int i = 0; i < 16; ++i) {
        const int h = sub * 16 + i;
        const float val = px[h] + pa[h] / pd[h >> 5] + gat_b[h];
        r[i] = val; s += val; s2 += val * val;
    }
    #pragma unroll
    for (int off = 4; off >= 1; off >>= 1) {
        s  += __shfl_xor(s,  off, 32);
        s2 += __shfl_xor(s2, off, 32);
    }
    const float mean = s * (1.0f / HH);
    const float inv  = rsqrtf(s2 * (1.0f / HH) - mean * mean + 1e-5f);
    float* op = out + (size_t)row * HH;
    #pragma unroll
    for (int i = 0; i < 16; ++i) {
        const int h = sub * 16 + i;
        op[h] = (r[i] - mean) * inv * ln2_g[h] + ln2_b[h];
    }
}

// ---------------------------------------------------------------------------
extern "C" void kernel_launch(void* const* d_in, const int* in_sizes, int n_in,
                              void* d_out, int out_size, void* d_ws, size_t ws_size,
                              hipStream_t stream)
{
    const float* x         = (const float*)d_in[0];
    const int*   ei        = (const int*)  d_in[1];
    const float* edge_attr = (const float*)d_in[2];
    const float* conv_w    = (const float*)d_in[3];
    const float* conv_b    = (const float*)d_in[4];
    const float* ln1_g     = (const float*)d_in[5];
    const float* ln1_b     = (const float*)d_in[6];
    const float* Wl        = (const float*)d_in[7];
    const float* Wr        = (const float*)d_in[8];
    const float* We        = (const float*)d_in[9];
    const float* att       = (const float*)d_in[10];
    const float* gat_b     = (const float*)d_in[11];
    const float* ln2_g     = (const float*)d_in[12];
    const float* ln2_b     = (const float*)d_in[13];
    float* out = (float*)d_out;

    const int Nn = in_sizes[0] / (TT * HH);   // nodes (1000)
    const int E  = in_sizes[2] / 16;          // edges (16000)
    const int R  = Nn * TT;                   // (node,t) rows (32000)

    // workspace layout (floats)
    float* ws = (float*)d_ws;
    size_t o = 0;
    float* x1     = ws + o; o += (size_t)R * HH;
    float* gl     = ws + o; o += (size_t)R * HH;
    float* gr     = ws + o; o += (size_t)R * HH;
    float* ee     = ws + o; o += (size_t)(E + Nn) * HH;
    float* logits = ws + o; o += (size_t)(E + Nn) * 128;
    float* zbase  = ws + o;                       // --- zero-initialized region ---
    unsigned* mx  = (unsigned*)(ws + o); o += (size_t)R * 4;
    float* den    = ws + o; o += (size_t)R * 4;
    float* aggr   = ws + o; o += (size_t)R * HH;
    unsigned* cnt = (unsigned*)(ws + o); o += (size_t)Nn;
    float* esum   = ws + o; o += (size_t)Nn * 16;
    const size_t zbytes = (size_t)((ws + o) - zbase) * sizeof(float);
    hipMemsetAsync(zbase, 0, zbytes, stream);     // mx=0 sorts below f2ord(-inf)

    k_conv_ln1<<<Nn, 256, 0, stream>>>(x, conv_w, conv_b, ln1_g, ln1_b, x1);
    k_gat_transform<<<R / 16, 256, 0, stream>>>(x1, Wl, Wr, gl, gr, Nn);
    k_loop_sum<<<(E * 16 + 255) / 256, 256, 0, stream>>>(ei, edge_attr, esum, cnt, E);
    const int egk = (E + Nn) * 128;
    k_edge_transform<<<(egk + 255) / 256, 256, 0, stream>>>(edge_attr, esum, cnt, We, ee, E, Nn);
    k_logits<<<(egk + 255) / 256, 256, 0, stream>>>(ei, gl, gr, ee, att, logits, mx, E, Nn);
    k_aggregate<<<(egk + 255) / 256, 256, 0, stream>>>(ei, gl, logits, mx, den, aggr, E, Nn);
    k_final_ln2<<<(R * 8 + 255) / 256, 256, 0, stream>>>(x1, aggr, den, gat_b, ln2_g, ln2_b, out, R);
}